// OuterProduct_652835029563
// MI455X (gfx1250) — compile-verified
//
#include <hip/hip_runtime.h>

typedef float v2f __attribute__((ext_vector_type(2)));
typedef float v8f __attribute__((ext_vector_type(8)));

#if defined(__HIP_DEVICE_COMPILE__) && !__has_builtin(__builtin_amdgcn_wmma_f32_16x16x4_f32)
#error "device pass: missing __builtin_amdgcn_wmma_f32_16x16x4_f32"
#endif

#define LQ   512
#define CIN  256
#define HID  32
#define PAIRD 64

// ---------------------------------------------------------------------------
// Kernel 1: s[i,d] = dot(seq[i,:], W1[d,:]) + b1[d]        (512 x 32, K=256)
// ---------------------------------------------------------------------------
__global__ void proj_kernel(const float* __restrict__ seq,
                            const float* __restrict__ W1,
                            const float* __restrict__ b1,
                            float* __restrict__ s) {
  int idx = blockIdx.x * blockDim.x + threadIdx.x;   // L*HID = 16384
  int i = idx >> 5, d = idx & 31;
  const float4* a = (const float4*)(seq + i * CIN);
  const float4* w = (const float4*)(W1 + d * CIN);
  float acc = b1[d];
#pragma unroll 8
  for (int t = 0; t < CIN / 4; ++t) {
    float4 av = a[t], wv = w[t];
    acc += av.x * wv.x + av.y * wv.y + av.z * wv.z + av.w * wv.w;
  }
  s[idx] = acc;
}

// ---------------------------------------------------------------------------
// Kernel 2: tmp[i,k,c] = dot(s[i,0:32], W2[k, c*32 : c*32+32])
// tmp layout: [i][k][c], 512*64*32 elements (4 MiB -> lives in L2)
// ---------------------------------------------------------------------------
__global__ void tmp_kernel(const float* __restrict__ s,
                           const float* __restrict__ W2,
                           float* __restrict__ tmp) {
  int idx = blockIdx.x * blockDim.x + threadIdx.x;   // 1,048,576
  int c = idx & 31;
  int k = (idx >> 5) & 63;
  int i = idx >> 11;
  const float4* sv = (const float4*)(s + i * HID);
  const float4* wv = (const float4*)(W2 + (k * HID + c) * HID);
  float acc = 0.0f;
#pragma unroll
  for (int t = 0; t < HID / 4; ++t) {
    float4 av = sv[t], bv = wv[t];
    acc += av.x * bv.x + av.y * bv.y + av.z * bv.z + av.w * bv.w;
  }
  tmp[idx] = acc;
}

// ---------------------------------------------------------------------------
// Kernel 3 (the 1.07 GFLOP one, memory-bound):
//   out[i,j,k] = sum_c tmp[i,k,c]*s[j,c] + b2[k] + pair[i,j,k]
// One wave per 16x16 (i,j) tile x 4 consecutive k.
// A = tmp tile [16 x 32] (M x K), B = s tile viewed as [32 x 16] (K x N);
// s row-major IS B^T row-major, so per-lane float2 loads match the ISA
// operand layout exactly:  lane l: A[l&15][2*(l>>4)+v], B[2*(l>>4)+v][l&15].
// ---------------------------------------------------------------------------
__global__ void outer_kernel(const float* __restrict__ s,
                             const float* __restrict__ tmp,
                             const float* __restrict__ pair,
                             const float* __restrict__ b2,
                             float* __restrict__ out) {
  const int lane = threadIdx.x & 31;
  const int wid  = blockIdx.x * 8 + (threadIdx.x >> 5);  // 8 waves / block
  // wid in [0, 32*32*16): k-group fastest -> waves in a block share pair lines
  const int kg = wid & 15;
  const int tj = (wid >> 4) & 31;
  const int ti = wid >> 9;
  const int k0 = kg << 2;
  const int i0 = ti << 4;
  const int j0 = tj << 4;
  const int l16 = lane & 15;
  const int h   = lane >> 4;

  // Prefetch the epilogue pair_rep lines while the WMMAs run.
#pragma unroll
  for (int r = 0; r < 8; ++r) {
    int row = i0 + r + 8 * h;
    int idx = ((row << 9) + (j0 + l16)) * PAIRD + k0;
    __builtin_prefetch(pair + idx, 0, 1);
  }

  const float* srow = s   + (j0 + l16) * HID + 2 * h;
  const float* arow = tmp + ((i0 + l16) * PAIRD + k0) * HID + 2 * h;

  v8f acc0 = {}, acc1 = {}, acc2 = {}, acc3 = {};
#pragma unroll
  for (int step = 0; step < 8; ++step) {
    const int c0 = step * 4;
    v2f b  = *(const v2f*)(srow + c0);
    v2f a0 = *(const v2f*)(arow + 0 * HID + c0);
    v2f a1 = *(const v2f*)(arow + 1 * HID + c0);
    v2f a2 = *(const v2f*)(arow + 2 * HID + c0);
    v2f a3 = *(const v2f*)(arow + 3 * HID + c0);
#if defined(__HIP_DEVICE_COMPILE__)
    acc0 = __builtin_amdgcn_wmma_f32_16x16x4_f32(false, a0, false, b, (short)0, acc0, false, false);
    acc1 = __builtin_amdgcn_wmma_f32_16x16x4_f32(false, a1, false, b, (short)0, acc1, false, false);
    acc2 = __builtin_amdgcn_wmma_f32_16x16x4_f32(false, a2, false, b, (short)0, acc2, false, false);
    acc3 = __builtin_amdgcn_wmma_f32_16x16x4_f32(false, a3, false, b, (short)0, acc3, false, false);
#else
    (void)a0; (void)a1; (void)a2; (void)a3; (void)b;
#endif
  }

  const float4 bias = *(const float4*)(b2 + k0);
  // D layout: lane l, acc element r  ->  out[i0 + r + 8h][j0 + (l&15)][k0..k0+3]
#pragma unroll
  for (int r = 0; r < 8; ++r) {
    int row = i0 + r + 8 * h;
    int idx = ((row << 9) + (j0 + l16)) * PAIRD + k0;   // < 2^24, fits int
    float4 p = *(const float4*)(pair + idx);
    float4 o;
    o.x = acc0[r] + bias.x + p.x;
    o.y = acc1[r] + bias.y + p.y;
    o.z = acc2[r] + bias.z + p.z;
    o.w = acc3[r] + bias.w + p.w;
    *(float4*)(out + idx) = o;
  }
}

// ---------------------------------------------------------------------------
extern "C" void kernel_launch(void* const* d_in, const int* in_sizes, int n_in,
                              void* d_out, int out_size, void* d_ws, size_t ws_size,
                              hipStream_t stream) {
  const float* seq  = (const float*)d_in[0];  // [1, 512, 256]
  const float* pair = (const float*)d_in[1];  // [1, 512, 512, 64]
  const float* W1   = (const float*)d_in[2];  // [32, 256]
  const float* b1   = (const float*)d_in[3];  // [32]
  const float* W2   = (const float*)d_in[4];  // [64, 1024]
  const float* b2   = (const float*)d_in[5];  // [64]
  float* out = (float*)d_out;                 // [1, 512, 512, 64]

  float* s   = (float*)d_ws;                        // 512*32   floats (64 KB)
  float* tmp = (float*)((char*)d_ws + 64 * 1024);   // 512*64*32 floats (4 MiB)

  proj_kernel<<<(LQ * HID) / 256, 256, 0, stream>>>(seq, W1, b1, s);
  tmp_kernel<<<(LQ * PAIRD * HID) / 256, 256, 0, stream>>>(s, W2, tmp);
  // 32*32 tiles * 16 k-groups = 16384 waves; 8 waves/block -> 2048 blocks
  outer_kernel<<<2048, 256, 0, stream>>>(s, tmp, pair, b2, out);
}